// CorrelatedAttentionBlock_42391327211983
// MI455X (gfx1250) — compile-verified
//
#include <hip/hip_runtime.h>
#include <hip/hip_bf16.h>

typedef __attribute__((ext_vector_type(2))) float v2f;
typedef __attribute__((ext_vector_type(8))) float v8f;

#define B_   4
#define T_   512
#define D_   512
#define H_   8
#define DH   64
#define L_   64
#define KTOP 18
#define NJ   (KTOP + 1)   // slot 0 = lag-0 (instant path), 1..18 = top lags

__device__ __forceinline__ int lag_of(int l) { return (l == 63) ? 511 : (1 + 7 * l); }

__device__ __forceinline__ v8f wmma_f32(v2f a, v2f b, v8f c) {
  // V_WMMA_F32_16X16X4_F32 : D(16x16 f32) = A(16x4 f32) * B(4x16 f32) + C
  return __builtin_amdgcn_wmma_f32_16x16x4_f32(false, a, false, b, (short)0, c, false, false);
}

// ---------------------------------------------------------------------------
// GEMM: Y = X(2048x512) @ W(512x512) + bias ; optional head-split store
// one wave per 16x16 output tile
// ---------------------------------------------------------------------------
__global__ void k_gemm_proj(const float* __restrict__ X, const float* __restrict__ W,
                            const float* __restrict__ bias, float* __restrict__ Y,
                            int headSplit) {
  const int lane = threadIdx.x;          // block = 32 (one wave)
  const int l16  = lane & 15;
  const int half = lane >> 4;
  const int m0 = blockIdx.x * 16;        // row tile over B*T
  const int n0 = blockIdx.y * 16;        // col tile over D
  v8f acc = {};
  for (int k0 = 0; k0 < D_; k0 += 4) {
    const int ka = k0 + 2 * half;
    const float* xr = X + (size_t)(m0 + l16) * D_ + ka;
    v2f a; a.x = xr[0]; a.y = xr[1];
    v2f b; b.x = W[(size_t)ka * D_ + n0 + l16];
    b.y = W[(size_t)(ka + 1) * D_ + n0 + l16];
    acc = wmma_f32(a, b, acc);
  }
  const int col = n0 + l16;
  const float bv = bias[col];
  for (int r = 0; r < 8; ++r) {
    const int row = m0 + r + 8 * half;
    const float v = acc[r] + bv;
    if (headSplit) {
      const int bb = row >> 9, t = row & (T_ - 1);
      const int h = col >> 6, d = col & (DH - 1);
      Y[(((size_t)(bb * H_ + h)) * T_ + t) * DH + d] = v;
    } else {
      Y[(size_t)row * D_ + col] = v;
    }
  }
}

// ---------------------------------------------------------------------------
// L2 normalize Q and K over the time axis: per (b,h,d) column of length T
// blockIdx < 2048 -> Q column, else K column
// ---------------------------------------------------------------------------
__global__ void k_l2norm_time(float* __restrict__ Q, float* __restrict__ K) {
  __shared__ float red[256];
  const int bid = blockIdx.x;
  float* P = (bid < 2048) ? Q : K;
  const int col = bid & 2047;
  const int bh = col >> 6, d = col & 63;
  float* base = P + (size_t)bh * T_ * DH + d;
  const int tid = threadIdx.x;           // 256
  const float v0 = base[(size_t)tid * DH];
  const float v1 = base[(size_t)(tid + 256) * DH];
  red[tid] = v0 * v0 + v1 * v1;
  __syncthreads();
  for (int off = 128; off > 0; off >>= 1) {
    if (tid < off) red[tid] += red[tid + off];
    __syncthreads();
  }
  const float inv = 1.0f / sqrtf(fmaxf(red[0], 1e-8f));
  base[(size_t)tid * DH] = v0 * inv;
  base[(size_t)(tid + 256) * DH] = v1 * inv;
}

// ---------------------------------------------------------------------------
// cov tile engine: acc[ct] = tile (dt,ct) of Kroll^T(64xT) @ Qhat(Tx64)
// ---------------------------------------------------------------------------
__device__ __forceinline__ void cov_tiles(const float* __restrict__ Kp,
                                          const float* __restrict__ Qp,
                                          int lag, int dt, int l16, int half,
                                          v8f acc[4]) {
  for (int t0 = 0; t0 < T_; t0 += 4) {
    const int ta = t0 + 2 * half;
    int s0 = ta - lag;     if (s0 < 0) s0 += T_;
    int s1 = ta + 1 - lag; if (s1 < 0) s1 += T_;
    v2f a;
    a.x = Kp[(size_t)s0 * DH + dt * 16 + l16];
    a.y = Kp[(size_t)s1 * DH + dt * 16 + l16];
    const float* q0 = Qp + (size_t)ta * DH;
    const float* q1 = Qp + (size_t)(ta + 1) * DH;
    for (int ct = 0; ct < 4; ++ct) {
      v2f b; b.x = q0[ct * 16 + l16]; b.y = q1[ct * 16 + l16];
      acc[ct] = wmma_f32(a, b, acc[ct]);
    }
  }
}

// ---------------------------------------------------------------------------
// score[b,h,l] = lam*sum|diag(cov)| + (1-lam)*(sum|cov| - sum|diag|)
// one block (4 waves) per (bh, lag); cov never materialized
// ---------------------------------------------------------------------------
__global__ void k_cov_score(const float* __restrict__ Khat, const float* __restrict__ Qhat,
                            const float* __restrict__ lam_p, float* __restrict__ score) {
  __shared__ float rtot[128], rdiag[128];
  const int bh = blockIdx.x;
  const int l  = blockIdx.y;
  const int lag = lag_of(l);
  const int tid = threadIdx.x;           // 128
  const int lane = tid & 31, wave = tid >> 5;
  const int l16 = lane & 15, half = lane >> 4;
  const float* Kp = Khat + (size_t)bh * T_ * DH;
  const float* Qp = Qhat + (size_t)bh * T_ * DH;
  v8f acc[4] = {};
  cov_tiles(Kp, Qp, lag, wave, l16, half, acc);
  float tot = 0.f, dg = 0.f;
  for (int ct = 0; ct < 4; ++ct)
    for (int r = 0; r < 8; ++r) {
      const float v = fabsf(acc[ct][r]);
      tot += v;
      const int d = wave * 16 + r + 8 * half;
      const int c = ct * 16 + l16;
      if (d == c) dg += v;
    }
  rtot[tid] = tot; rdiag[tid] = dg;
  __syncthreads();
  for (int off = 64; off > 0; off >>= 1) {
    if (tid < off) { rtot[tid] += rtot[tid + off]; rdiag[tid] += rdiag[tid + off]; }
    __syncthreads();
  }
  if (tid == 0) {
    const float lam = fminf(fmaxf(lam_p[0], 0.f), 1.f);
    const float dd = rdiag[0], tt = rtot[0];
    score[bh * L_ + l] = lam * dd + (1.f - lam) * (tt - dd);
  }
}

// ---------------------------------------------------------------------------
// top-18 lag selection (stable, descending) + softmax(top_scores/tau_lag)
// ---------------------------------------------------------------------------
__global__ void k_topk(const float* __restrict__ score, const float* __restrict__ logtau_lag,
                       int* __restrict__ topidx, float* __restrict__ wts) {
  const int bh = blockIdx.x;
  if (threadIdx.x != 0) return;
  float s[L_];
  for (int i = 0; i < L_; ++i) s[i] = score[bh * L_ + i];
  int idx[KTOP]; float val[KTOP];
  for (int k = 0; k < KTOP; ++k) {
    int best = 0; float bv = s[0];
    for (int i = 1; i < L_; ++i)
      if (s[i] > bv) { bv = s[i]; best = i; }
    idx[k] = best; val[k] = bv; s[best] = -3.4e38f;
  }
  const float tl = fmaxf(expf(logtau_lag[0]), 1e-4f);
  const float m = val[0] / tl;           // sorted descending -> val[0] is max
  float sum = 0.f, e[KTOP];
  for (int k = 0; k < KTOP; ++k) { e[k] = expf(val[k] / tl - m); sum += e[k]; }
  for (int k = 0; k < KTOP; ++k) {
    topidx[bh * KTOP + k] = idx[k];
    wts[bh * KTOP + k] = e[k] / sum;
  }
}

// ---------------------------------------------------------------------------
// materialize cov for lag-0 (j=0) and the 18 selected lags (j=1..18)
// ---------------------------------------------------------------------------
__global__ void k_cov_top(const float* __restrict__ Khat, const float* __restrict__ Qhat,
                          const int* __restrict__ topidx, float* __restrict__ covbuf) {
  const int bh = blockIdx.x;
  const int j  = blockIdx.y;             // 0..18
  const int lag = (j == 0) ? 0 : lag_of(topidx[bh * KTOP + (j - 1)]);
  const int tid = threadIdx.x;           // 128
  const int lane = tid & 31, wave = tid >> 5;
  const int l16 = lane & 15, half = lane >> 4;
  const float* Kp = Khat + (size_t)bh * T_ * DH;
  const float* Qp = Qhat + (size_t)bh * T_ * DH;
  v8f acc[4] = {};
  cov_tiles(Kp, Qp, lag, wave, l16, half, acc);
  float* out = covbuf + ((size_t)bh * NJ + j) * DH * DH;
  for (int ct = 0; ct < 4; ++ct)
    for (int r = 0; r < 8; ++r) {
      const int d = wave * 16 + r + 8 * half;
      const int c = ct * 16 + l16;
      out[(size_t)d * DH + c] = acc[ct][r];
    }
}

// ---------------------------------------------------------------------------
// in-place row softmax of cov/tau over last axis (rows of length 64)
// ---------------------------------------------------------------------------
__global__ void k_softmax_rows(float* __restrict__ covbuf, const float* __restrict__ logtau) {
  __shared__ float red[64];
  float* row = covbuf + (size_t)blockIdx.x * DH;
  const int tid = threadIdx.x;           // 64
  const float inv_tau = 1.0f / fmaxf(expf(logtau[0]), 1e-4f);
  const float y = row[tid] * inv_tau;
  red[tid] = y;
  __syncthreads();
  for (int off = 32; off > 0; off >>= 1) {
    if (tid < off) red[tid] = fmaxf(red[tid], red[tid + off]);
    __syncthreads();
  }
  const float m = red[0];
  __syncthreads();
  const float e = expf(y - m);
  red[tid] = e;
  __syncthreads();
  for (int off = 32; off > 0; off >>= 1) {
    if (tid < off) red[tid] += red[tid + off];
    __syncthreads();
  }
  row[tid] = e / red[0];
}

// ---------------------------------------------------------------------------
// out_h = (1-beta) * V@att0 + beta * sum_k w_k * roll(V,lag_k)@att_k
// one wave per 16x16 output tile, 19 WMMA mini-GEMMs per tile
// ---------------------------------------------------------------------------
__global__ void k_combine(const float* __restrict__ Vh, const float* __restrict__ covbuf,
                          const int* __restrict__ topidx, const float* __restrict__ wts,
                          const float* __restrict__ beta_p, float* __restrict__ outh) {
  const int lane = threadIdx.x;          // 32
  const int l16 = lane & 15, half = lane >> 4;
  const int bh = blockIdx.x;             // 32
  const int tile = blockIdx.y;           // 128 = 32 t-tiles * 4 c-tiles
  const int tt = tile >> 2, ct = tile & 3;
  const float* Vp = Vh + (size_t)bh * T_ * DH;
  const float* attb = covbuf + (size_t)bh * NJ * DH * DH;
  const float beta = fminf(fmaxf(beta_p[0], 0.f), 1.f);
  v8f total = {};
  for (int j = 0; j < NJ; ++j) {
    const int lag = (j == 0) ? 0 : lag_of(topidx[bh * KTOP + j - 1]);
    const float w = (j == 0) ? (1.f - beta) : (beta * wts[bh * KTOP + j - 1]);
    const float* att = attb + (size_t)j * DH * DH;
    int srow = tt * 16 + l16 - lag; if (srow < 0) srow += T_;
    const float* arow = Vp + (size_t)srow * DH;
    v8f acc = {};
    for (int k0 = 0; k0 < DH; k0 += 4) {
      const int ka = k0 + 2 * half;
      v2f a; a.x = arow[ka]; a.y = arow[ka + 1];
      v2f b; b.x = att[(size_t)ka * DH + ct * 16 + l16];
      b.y = att[(size_t)(ka + 1) * DH + ct * 16 + l16];
      acc = wmma_f32(a, b, acc);
    }
    for (int r = 0; r < 8; ++r) total[r] += w * acc[r];
  }
  const int b = bh >> 3, h = bh & 7;
  for (int r = 0; r < 8; ++r) {
    const int t = tt * 16 + r + 8 * half;
    const int c = ct * 16 + l16;
    outh[((size_t)b * T_ + t) * D_ + h * DH + c] = total[r];
  }
}

// ---------------------------------------------------------------------------
extern "C" void kernel_launch(void* const* d_in, const int* in_sizes, int n_in,
                              void* d_out, int out_size, void* d_ws, size_t ws_size,
                              hipStream_t stream) {
  const float* x   = (const float*)d_in[0];
  const float* Wq  = (const float*)d_in[1];
  const float* bq  = (const float*)d_in[2];
  const float* Wk  = (const float*)d_in[3];
  const float* bk  = (const float*)d_in[4];
  const float* Wv  = (const float*)d_in[5];
  const float* bv  = (const float*)d_in[6];
  const float* Wo  = (const float*)d_in[7];
  const float* bo  = (const float*)d_in[8];
  const float* log_tau     = (const float*)d_in[9];
  const float* lambda_auto = (const float*)d_in[10];
  const float* beta_lag    = (const float*)d_in[11];
  const float* log_tau_lag = (const float*)d_in[12];

  // workspace carve-up (floats)
  float* ws = (float*)d_ws;
  const size_t NQ = (size_t)B_ * H_ * T_ * DH;      // 1,048,576
  float* Qh = ws;
  float* Kh = Qh + NQ;
  float* Vh = Kh + NQ;
  float* scorebuf = Vh + NQ;                        // B*H*L = 2048
  int*   topidx = (int*)(scorebuf + (size_t)B_ * H_ * L_);
  float* wts = (float*)(topidx + (size_t)B_ * H_ * KTOP);
  float* covbuf = wts + (size_t)B_ * H_ * KTOP;     // 32*19*4096 floats
  float* outh = covbuf + (size_t)B_ * H_ * NJ * DH * DH;

  const dim3 gGemm(128, 32);   // (B*T/16, D/16)
  k_gemm_proj<<<gGemm, 32, 0, stream>>>(x, Wq, bq, Qh, 1);
  k_gemm_proj<<<gGemm, 32, 0, stream>>>(x, Wk, bk, Kh, 1);
  k_gemm_proj<<<gGemm, 32, 0, stream>>>(x, Wv, bv, Vh, 1);

  k_l2norm_time<<<2 * B_ * H_ * DH, 256, 0, stream>>>(Qh, Kh);

  k_cov_score<<<dim3(B_ * H_, L_), 128, 0, stream>>>(Kh, Qh, lambda_auto, scorebuf);
  k_topk<<<B_ * H_, 1, 0, stream>>>(scorebuf, log_tau_lag, topidx, wts);
  k_cov_top<<<dim3(B_ * H_, NJ), 128, 0, stream>>>(Kh, Qh, topidx, covbuf);
  k_softmax_rows<<<B_ * H_ * NJ * DH, 64, 0, stream>>>(covbuf, log_tau);

  k_combine<<<dim3(B_ * H_, 128), 32, 0, stream>>>(Vh, covbuf, topidx, wts, beta_lag, outh);

  k_gemm_proj<<<gGemm, 32, 0, stream>>>(outh, Wo, bo, (float*)d_out, 0);
}